// SprVin_64037962383688
// MI455X (gfx1250) — compile-verified
//
#include <hip/hip_runtime.h>
#include <hip/hip_bf16.h>

typedef __attribute__((ext_vector_type(2))) float v2f;
typedef __attribute__((ext_vector_type(8))) float v8f;

#define NB      128
#define HW      64
#define NPIX    4096          // 64*64
#define STRIDE  66            // 64 + 2 halo
#define PLANE   (STRIDE*STRIDE)
#define R_OFF   0
#define V0_OFF  PLANE
#define V1_OFF  (2*PLANE)
#define SCR_OFF (3*PLANE)
// +160 floats of zeroed padding so the pipeline's one dead over-prefetch
// (tile index 32, discarded) stays inside the LDS allocation.
#define LDS_FLOATS (3*PLANE + 160)
#define K_ITERS 30

// ---------------------------------------------------------------------------
// Kernel A: fuse conv150 + 1x1 conv into a single 3-in/1-out 3x3 conv.
// W_eff[ci*9+t] = sum_c r_w[c]*h_w[c,ci,t] ; b_eff = sum_c r_w[c]*h_b[c]
// ---------------------------------------------------------------------------
__global__ void vin_prep_kernel(const float* __restrict__ h_w,
                                const float* __restrict__ h_b,
                                const float* __restrict__ r_w,
                                float* __restrict__ weff) {
  int t = threadIdx.x;
  if (t < 27) {
    float s = 0.f;
    for (int c = 0; c < 150; ++c) s += r_w[c] * h_w[c * 27 + t];
    weff[t] = s;
  } else if (t == 27) {
    float s = 0.f;
    for (int c = 0; c < 150; ++c) s += r_w[c] * h_b[c];
    weff[27] = s;
  }
}

// Cross-half (xor 16) max of a float within a wave32 (pure VALU).
__device__ __forceinline__ float xhalf_max(float m) {
  int mi = __float_as_int(m);
  int oi = __builtin_amdgcn_permlanex16(mi, mi, 0x76543210, 0xfedcba98,
                                        false, false);
  return fmaxf(m, __int_as_float(oi));
}

// ---------------------------------------------------------------------------
// Kernel B: one persistent workgroup per image; whole VIN in LDS.
// ---------------------------------------------------------------------------
__global__ __launch_bounds__(256)
void vin_iter_kernel(const float* __restrict__ layout,   // (B,2,64,64)
                     const float* __restrict__ heat,     // (B,1,64,64)
                     const float* __restrict__ qw,       // (16,1,3,3)
                     const float* __restrict__ ww,       // (16,1,3,3)
                     const float* __restrict__ fcw,      // (4,16)
                     const int*   __restrict__ S1,
                     const int*   __restrict__ S2,
                     const float* __restrict__ weff,     // 27 + 1 fused wts
                     float* __restrict__ o_logits,
                     float* __restrict__ o_probs,
                     float* __restrict__ o_v,
                     float* __restrict__ o_r,
                     float* __restrict__ o_hm) {
  __shared__ float lds[LDS_FLOATS];
  const int b    = blockIdx.x;
  const int tid  = threadIdx.x;
  const int lane = tid & 31;
  const int wv   = tid >> 5;        // wave id 0..7
  const int half = lane >> 4;       // lane half: selects K pair within frag
  const int ch   = lane & 15;       // A: M=channel ; B: N=pixel-in-tile

  // Zero LDS: zero halos, v0 == 0 (iter 0 == conv(r,q_w)), zero pad region.
  for (int i = tid; i < LDS_FLOATS; i += 256) lds[i] = 0.f;

  float wef[27];
#pragma unroll
  for (int i = 0; i < 27; ++i) wef[i] = weff[i];
  const float beff = weff[27];
  __syncthreads();

  // ---- r = conv3x3(X, W_eff) + b_eff ; also emit r and heatmap outputs ----
  const float* xl = layout + (size_t)b * 2 * NPIX;
  const float* xh = heat   + (size_t)b * NPIX;
  for (int i = tid; i < NPIX; i += 256) {
    int y = i >> 6, x = i & 63;
    float acc = beff;
#pragma unroll
    for (int t = 0; t < 9; ++t) {
      int yy = y + t / 3 - 1, xx = x + t % 3 - 1;
      if (yy >= 0 && yy < HW && xx >= 0 && xx < HW) {
        int off = yy * HW + xx;
        acc += wef[t]      * xl[off];
        acc += wef[9 + t]  * xl[NPIX + off];
        acc += wef[18 + t] * xh[off];
      }
    }
    lds[R_OFF + (y + 1) * STRIDE + (x + 1)] = acc;
    o_r[(size_t)b * NPIX + i]  = acc;
    o_hm[(size_t)b * NPIX + i] = xh[i];
  }

  // ---- A fragments: wq = [q_w | w], M=channel(16), K=18 padded to 20 ------
  // f32 16x4 A layout: lanes 0-15 hold K={4j,4j+1}, lanes 16-31 K={4j+2,4j+3}
  v2f afrag[5];
#pragma unroll
  for (int j = 0; j < 5; ++j) {
    int k0 = 4 * j + 2 * half;
    int k1 = k0 + 1;
    float a0 = (k0 < 9) ? qw[ch * 9 + k0] : ((k0 < 18) ? ww[ch * 9 + (k0 - 9)] : 0.f);
    float a1 = (k1 < 9) ? qw[ch * 9 + k1] : ((k1 < 18) ? ww[ch * 9 + (k1 - 9)] : 0.f);
    afrag[j].x = a0;
    afrag[j].y = a1;
  }

  // ---- Per-lane B-tap induction variables ---------------------------------
  // Lane's pixel column is constant; the wave's row advances by 2 per tile.
  // k>=18 lanes are clamped to k=17: their A-weight is exactly 0, so the
  // (finite) clamped tap contributes 0.
  const int y0   = wv >> 2;                  // starting row of this wave
  const int xcol = ((wv & 3) << 4) + ch;     // this lane's pixel column
  int toff[10];
  int tisv[10];
#pragma unroll
  for (int u = 0; u < 10; ++u) {
    int k = 4 * (u >> 1) + 2 * half + (u & 1);
    if (k > 17) k = 17;
    int isv = (k >= 9) ? 1 : 0;
    int t = isv ? (k - 9) : k;
    toff[u] = (t / 3) * STRIDE + (t % 3) + xcol;   // tap addr at row y=0
    tisv[u] = isv;
  }
  __syncthreads();

  // ---- value iteration: v <- max_ch conv3x3([r, v], wq), 30 updates -------
#pragma unroll 1
  for (int it = 0; it < K_ITERS; ++it) {
    const int vc = (it & 1) ? V1_OFF : V0_OFF;   // read plane
    const int vn = (it & 1) ? V0_OFF : V1_OFF;   // write plane

    int addr[10];
#pragma unroll
    for (int u = 0; u < 10; ++u)
      addr[u] = toff[u] + (tisv[u] ? vc : 0) + y0 * STRIDE;
    int saddr = vn + (y0 + 1) * STRIDE + (xcol + 1);

    // Ping-pong pipeline, 2 tiles per body: taps for tile n+1 load while
    // tile n runs its WMMAs. Two independent accumulator chains (3+2 WMMAs)
    // merged in the max-reduce to shorten the dependent matrix-pipe chain.
    float ba[10], bb[10];
#pragma unroll
    for (int u = 0; u < 10; ++u) ba[u] = lds[addr[u]];

#pragma unroll 1
    for (int t8 = 0; t8 < 32; t8 += 2) {
      // prefetch taps for tile t8+1
#pragma unroll
      for (int u = 0; u < 10; ++u) {
        addr[u] += 2 * STRIDE;
        bb[u] = lds[addr[u]];
      }
      {
        v2f b0 = {ba[0], ba[1]}, b1 = {ba[2], ba[3]}, b2 = {ba[4], ba[5]};
        v2f b3 = {ba[6], ba[7]}, b4 = {ba[8], ba[9]};
        v8f c0 = {0.f, 0.f, 0.f, 0.f, 0.f, 0.f, 0.f, 0.f};
        v8f c1 = {0.f, 0.f, 0.f, 0.f, 0.f, 0.f, 0.f, 0.f};
        c0 = __builtin_amdgcn_wmma_f32_16x16x4_f32(false, afrag[0], false, b0, (short)0, c0, false, false);
        c1 = __builtin_amdgcn_wmma_f32_16x16x4_f32(false, afrag[3], false, b3, (short)0, c1, false, false);
        c0 = __builtin_amdgcn_wmma_f32_16x16x4_f32(false, afrag[1], false, b1, (short)0, c0, false, false);
        c1 = __builtin_amdgcn_wmma_f32_16x16x4_f32(false, afrag[4], false, b4, (short)0, c1, false, false);
        c0 = __builtin_amdgcn_wmma_f32_16x16x4_f32(false, afrag[2], false, b2, (short)0, c0, false, false);
        float m = c0[0] + c1[0];
        m = fmaxf(m, c0[1] + c1[1]); m = fmaxf(m, c0[2] + c1[2]);
        m = fmaxf(m, c0[3] + c1[3]); m = fmaxf(m, c0[4] + c1[4]);
        m = fmaxf(m, c0[5] + c1[5]); m = fmaxf(m, c0[6] + c1[6]);
        m = fmaxf(m, c0[7] + c1[7]);
        m = xhalf_max(m);
        if (lane < 16) lds[saddr] = m;
      }
      // prefetch taps for tile t8+2 (last body over-reads zeroed pad; unused)
#pragma unroll
      for (int u = 0; u < 10; ++u) {
        addr[u] += 2 * STRIDE;
        ba[u] = lds[addr[u]];
      }
      {
        v2f b0 = {bb[0], bb[1]}, b1 = {bb[2], bb[3]}, b2 = {bb[4], bb[5]};
        v2f b3 = {bb[6], bb[7]}, b4 = {bb[8], bb[9]};
        v8f c0 = {0.f, 0.f, 0.f, 0.f, 0.f, 0.f, 0.f, 0.f};
        v8f c1 = {0.f, 0.f, 0.f, 0.f, 0.f, 0.f, 0.f, 0.f};
        c0 = __builtin_amdgcn_wmma_f32_16x16x4_f32(false, afrag[0], false, b0, (short)0, c0, false, false);
        c1 = __builtin_amdgcn_wmma_f32_16x16x4_f32(false, afrag[3], false, b3, (short)0, c1, false, false);
        c0 = __builtin_amdgcn_wmma_f32_16x16x4_f32(false, afrag[1], false, b1, (short)0, c0, false, false);
        c1 = __builtin_amdgcn_wmma_f32_16x16x4_f32(false, afrag[4], false, b4, (short)0, c1, false, false);
        c0 = __builtin_amdgcn_wmma_f32_16x16x4_f32(false, afrag[2], false, b2, (short)0, c0, false, false);
        float m = c0[0] + c1[0];
        m = fmaxf(m, c0[1] + c1[1]); m = fmaxf(m, c0[2] + c1[2]);
        m = fmaxf(m, c0[3] + c1[3]); m = fmaxf(m, c0[4] + c1[4]);
        m = fmaxf(m, c0[5] + c1[5]); m = fmaxf(m, c0[6] + c1[6]);
        m = fmaxf(m, c0[7] + c1[7]);
        m = xhalf_max(m);
        if (lane < 16) lds[saddr + 2 * STRIDE] = m;
      }
      saddr += 4 * STRIDE;
    }
    __syncthreads();
  }
  // After 30 updates the live v buffer is V0 (matches reference's returned v).

  // ---- final conv only at (S1,S2): 16 channels, then fc + softmax ---------
  const int s1 = S1[b], s2 = S2[b];
  if (tid < 16) {
    float q = 0.f;
#pragma unroll
    for (int k = 0; k < 18; ++k) {
      int isv = (k >= 9);
      int t = isv ? (k - 9) : k;
      float wgt = isv ? ww[tid * 9 + t] : qw[tid * 9 + t];
      q += wgt * lds[(isv ? V0_OFF : R_OFF) + (s1 + t / 3) * STRIDE + (s2 + t % 3)];
    }
    lds[SCR_OFF + tid] = q;
  }
  __syncthreads();
  if (tid == 0) {
    float l[4];
#pragma unroll
    for (int j = 0; j < 4; ++j) {
      float s = 0.f;
#pragma unroll
      for (int c2 = 0; c2 < 16; ++c2) s += fcw[j * 16 + c2] * lds[SCR_OFF + c2];
      l[j] = s;
    }
    float mx = fmaxf(fmaxf(l[0], l[1]), fmaxf(l[2], l[3]));
    float e0 = expf(l[0] - mx), e1 = expf(l[1] - mx);
    float e2 = expf(l[2] - mx), e3 = expf(l[3] - mx);
    float inv = 1.f / (e0 + e1 + e2 + e3);
    o_logits[b * 4 + 0] = l[0]; o_logits[b * 4 + 1] = l[1];
    o_logits[b * 4 + 2] = l[2]; o_logits[b * 4 + 3] = l[3];
    o_probs[b * 4 + 0] = e0 * inv; o_probs[b * 4 + 1] = e1 * inv;
    o_probs[b * 4 + 2] = e2 * inv; o_probs[b * 4 + 3] = e3 * inv;
  }

  // ---- emit v ----
  for (int i = tid; i < NPIX; i += 256) {
    int y = i >> 6, x = i & 63;
    o_v[(size_t)b * NPIX + i] = lds[V0_OFF + (y + 1) * STRIDE + (x + 1)];
  }
}

// ---------------------------------------------------------------------------
extern "C" void kernel_launch(void* const* d_in, const int* in_sizes, int n_in,
                              void* d_out, int out_size, void* d_ws, size_t ws_size,
                              hipStream_t stream) {
  const float* layout = (const float*)d_in[0];
  const float* heat   = (const float*)d_in[1];
  const float* h_w    = (const float*)d_in[2];
  const float* h_b    = (const float*)d_in[3];
  const float* r_w    = (const float*)d_in[4];
  const float* q_w    = (const float*)d_in[5];
  const float* w      = (const float*)d_in[6];
  const float* fc_w   = (const float*)d_in[7];
  const int*   S1     = (const int*)d_in[8];
  const int*   S2     = (const int*)d_in[9];

  float* out = (float*)d_out;
  // Output layout (flat, return order): logits | probs | v | r | heatmap
  float* o_logits = out;
  float* o_probs  = out + NB * 4;
  float* o_v      = out + 2 * NB * 4;
  float* o_r      = o_v + NB * NPIX;
  float* o_hm     = o_r + NB * NPIX;

  float* weff = (float*)d_ws;  // 28 floats of fused first-layer weights

  vin_prep_kernel<<<1, 32, 0, stream>>>(h_w, h_b, r_w, weff);
  vin_iter_kernel<<<NB, 256, 0, stream>>>(layout, heat, q_w, w, fc_w, S1, S2,
                                          weff, o_logits, o_probs, o_v, o_r, o_hm);
}